// FDMM_72370198938104
// MI455X (gfx1250) — compile-verified
//
#include <hip/hip_runtime.h>

typedef float v2f __attribute__((ext_vector_type(2)));
typedef float v8f __attribute__((ext_vector_type(8)));

#define HW   16384
#define WID  128
#define BN   4
#define CN   32
#define MIDC 16

__device__ __forceinline__ float lrelu02(float v) { return v > 0.f ? v : 0.2f * v; }

// ---------------------------------------------------------------------------
// Kernel A: 1x1 conv over concat[x,y] (K=64 -> M=32) + bias + ReLU, via
// V_WMMA_F32_16X16X4_F32. One wave computes one 16(oc) x 16(pixel) tile.
// A layout (16x4 f32): v0 = K{0,2} by lane-half, v1 = K{1,3}; B mirrored.
// ---------------------------------------------------------------------------
__global__ void k_conv1x1_cat_relu(const float* __restrict__ x, const float* __restrict__ y,
                                   const float* __restrict__ w, const float* __restrict__ bias,
                                   float* __restrict__ out) {
  const int lane = threadIdx.x & 31;
  const int wv   = threadIdx.x >> 5;
  const int t    = blockIdx.x * (blockDim.x >> 5) + wv;   // 0..32767
  const int mt   = t & 1;
  const int nt   = t >> 1;
  const int gp   = nt << 4;              // global pixel over B*HW
  const int bb   = gp >> 14;
  const int p    = gp & (HW - 1);
  const int mbase = mt << 4;
  const int m    = mbase + (lane & 15);
  const int n    = lane & 15;
  const int kh   = lane >> 4;

  v8f acc = {0.f, 0.f, 0.f, 0.f, 0.f, 0.f, 0.f, 0.f};
#pragma unroll
  for (int k0 = 0; k0 < 64; k0 += 4) {
    const int ka = k0 + 2 * kh;          // even, never straddles the 32 boundary
    v2f av;
    av.x = w[m * 64 + ka];
    av.y = w[m * 64 + ka + 1];
    const float* src = (ka < 32) ? x : y;
    const int c  = ka & 31;
    const int ib = (bb * 32 + c) * HW + p + n;
    v2f bv;
    bv.x = src[ib];
    bv.y = src[ib + HW];
    acc = __builtin_amdgcn_wmma_f32_16x16x4_f32(false, av, false, bv, (short)0, acc, false, false);
  }
#pragma unroll
  for (int r = 0; r < 8; ++r) {
    const int oc = mbase + r + (kh << 3);
    float v = acc[r] + bias[oc];
    out[(bb * 32 + oc) * HW + p + n] = fmaxf(v, 0.f);
  }
}

// ---------------------------------------------------------------------------
// Kernel B: 3x3 conv C->C pad=1 + bias + ReLU as 9 shifted K=32 GEMMs,
// accumulated in one WMMA accumulator. Row clip is wave-uniform; column
// clip via per-lane select (EXEC stays all-ones around v_wmma).
// ---------------------------------------------------------------------------
__global__ void k_conv3x3_relu(const float* __restrict__ in, const float* __restrict__ w,
                               const float* __restrict__ bias, float* __restrict__ out) {
  const int lane = threadIdx.x & 31;
  const int wv   = threadIdx.x >> 5;
  const int t    = blockIdx.x * (blockDim.x >> 5) + wv;
  const int mt   = t & 1;
  const int nt   = t >> 1;
  const int gp   = nt << 4;
  const int bb   = gp >> 14;
  const int p    = gp & (HW - 1);
  const int yy   = p >> 7;
  const int x0   = p & (WID - 1);
  const int mbase = mt << 4;
  const int m    = mbase + (lane & 15);
  const int n    = lane & 15;
  const int kh   = lane >> 4;

  v8f acc = {0.f, 0.f, 0.f, 0.f, 0.f, 0.f, 0.f, 0.f};
  for (int dy = 0; dy < 3; ++dy) {
    const int iy = yy + dy - 1;
    if (iy < 0 || iy >= WID) continue;   // uniform across the wave (tile = one row)
    for (int dx = 0; dx < 3; ++dx) {
      const int ix = x0 + n + dx - 1;
      const bool ok = (ix >= 0) && (ix < WID);
      const int rb = (bb * 32) * HW + iy * WID + ix;
#pragma unroll
      for (int k0 = 0; k0 < 32; k0 += 4) {
        const int ka = k0 + 2 * kh;
        v2f av;
        av.x = w[((m * 32 + ka) * 3 + dy) * 3 + dx];
        av.y = w[((m * 32 + ka + 1) * 3 + dy) * 3 + dx];
        v2f bv;
        bv.x = ok ? in[rb + ka * HW] : 0.f;
        bv.y = ok ? in[rb + (ka + 1) * HW] : 0.f;
        acc = __builtin_amdgcn_wmma_f32_16x16x4_f32(false, av, false, bv, (short)0, acc, false, false);
      }
    }
  }
#pragma unroll
  for (int r = 0; r < 8; ++r) {
    const int oc = mbase + r + (kh << 3);
    float v = acc[r] + bias[oc];
    out[(bb * 32 + oc) * HW + p + n] = fmaxf(v, 0.f);
  }
}

// ---------------------------------------------------------------------------
// Kernel C: a2 = sigmoid(1x1 conv C->2), then xm = x*a2[0], ym = y*a2[1].
// ---------------------------------------------------------------------------
__global__ void k_attn_apply(const float* __restrict__ a, const float* __restrict__ w,
                             const float* __restrict__ bias,
                             const float* __restrict__ x, const float* __restrict__ y,
                             float* __restrict__ xm, float* __restrict__ ym) {
  const int idx = blockIdx.x * blockDim.x + threadIdx.x;
  if (idx >= BN * HW) return;
  const int p = idx & (HW - 1);
  const int b = idx >> 14;
  float s0 = bias[0], s1 = bias[1];
#pragma unroll
  for (int ic = 0; ic < CN; ++ic) {
    const float v = a[(b * CN + ic) * HW + p];
    s0 += w[ic] * v;
    s1 += w[CN + ic] * v;
  }
  s0 = 1.f / (1.f + expf(-s0));
  s1 = 1.f / (1.f + expf(-s1));
#pragma unroll
  for (int c = 0; c < CN; ++c) {
    const int o = (b * CN + c) * HW + p;
    xm[o] = x[o] * s0;
    ym[o] = y[o] * s1;
  }
}

// ---------------------------------------------------------------------------
// Depthwise conv (templated on k), bias, pad.
// ---------------------------------------------------------------------------
template <int K>
__global__ void k_dwconv(const float* __restrict__ in, const float* __restrict__ w,
                         const float* __restrict__ bias, float* __restrict__ out,
                         int nch, int pad) {
  const int idx = blockIdx.x * blockDim.x + threadIdx.x;
  if (idx >= BN * nch * HW) return;
  const int p  = idx & (HW - 1);
  const int bc = idx >> 14;
  const int ch = bc % nch;
  const int yy = p >> 7, xx = p & (WID - 1);
  float acc = bias[ch];
#pragma unroll
  for (int di = 0; di < K; ++di) {
    const int iy = yy + di - pad;
    if (iy < 0 || iy >= WID) continue;
#pragma unroll
    for (int dj = 0; dj < K; ++dj) {
      const int ix = xx + dj - pad;
      if (ix < 0 || ix >= WID) continue;
      acc += w[ch * K * K + di * K + dj] * in[bc * HW + iy * WID + ix];
    }
  }
  out[bc * HW + p] = acc;
}

// ---------------------------------------------------------------------------
// Pointwise 1x1 conv, runtime M/K, optional leaky-relu(0.2). Output batch
// stride is a parameter so pw3 can write straight into the (B,35,H,W) buffer.
// ---------------------------------------------------------------------------
__global__ void k_pwconv(const float* __restrict__ in, const float* __restrict__ w,
                         const float* __restrict__ bias, float* __restrict__ out,
                         int M, int K, int outBS, int doLrelu) {
  const int idx = blockIdx.x * blockDim.x + threadIdx.x;
  if (idx >= BN * M * HW) return;
  const int p  = idx & (HW - 1);
  const int bm = idx >> 14;
  const int m  = bm % M;
  const int b  = bm / M;
  float acc = bias[m];
  for (int kk = 0; kk < K; ++kk)
    acc += w[m * K + kk] * in[(b * K + kk) * HW + p];
  if (doLrelu) acc = lrelu02(acc);
  out[b * outBS + m * HW + p] = acc;
}

// ---------------------------------------------------------------------------
// Global average pool of xm over H,W: one block per (b,c).
// ---------------------------------------------------------------------------
__global__ void k_gap(const float* __restrict__ xm, float* __restrict__ gap) {
  __shared__ float red[256];
  const int bc = blockIdx.x;
  float s = 0.f;
  for (int i = threadIdx.x; i < HW; i += 256) s += xm[bc * HW + i];
  red[threadIdx.x] = s;
  __syncthreads();
  for (int st = 128; st > 0; st >>= 1) {
    if (threadIdx.x < st) red[threadIdx.x] += red[threadIdx.x + st];
    __syncthreads();
  }
  if (threadIdx.x == 0) gap[bc] = red[0] * (1.f / (float)HW);
}

// SE layer 1: (B, MID) = lrelu(W1 @ gap + b1)
__global__ void k_se1(const float* __restrict__ gap, const float* __restrict__ w,
                      const float* __restrict__ b1, float* __restrict__ g1) {
  const int tid = threadIdx.x;
  if (tid >= BN * MIDC) return;
  const int b = tid >> 4, m = tid & 15;
  float acc = b1[m];
#pragma unroll
  for (int ic = 0; ic < CN; ++ic) acc += w[m * CN + ic] * gap[b * CN + ic];
  g1[b * MIDC + m] = lrelu02(acc);
}

// SE layer 2: writes branch slice into spe (B, C, 35)
__global__ void k_se2(const float* __restrict__ g1, const float* __restrict__ w,
                      const float* __restrict__ b2, float* __restrict__ spe,
                      int k2, int off) {
  const int idx = blockIdx.x * blockDim.x + threadIdx.x;
  if (idx >= BN * CN * k2) return;
  const int b   = idx / (CN * k2);
  const int rem = idx % (CN * k2);
  const int c   = rem / k2;
  const int j   = rem % k2;
  float acc = b2[rem];
#pragma unroll
  for (int mid = 0; mid < MIDC; ++mid) acc += w[rem * MIDC + mid] * g1[b * MIDC + mid];
  spe[(b * CN + c) * 35 + off + j] = acc;
}

// KernelNorm over the 35 spatial channels (ddof=1), in place.
__global__ void k_norm_spa(float* __restrict__ spa) {
  const int idx = blockIdx.x * blockDim.x + threadIdx.x;
  if (idx >= BN * HW) return;
  const int b = idx >> 14, p = idx & (HW - 1);
  float v[35];
  float mean = 0.f;
#pragma unroll
  for (int ch = 0; ch < 35; ++ch) {
    v[ch] = spa[(b * 35 + ch) * HW + p];
    mean += v[ch];
  }
  mean *= (1.f / 35.f);
  float var = 0.f;
#pragma unroll
  for (int ch = 0; ch < 35; ++ch) {
    const float d = v[ch] - mean;
    var += d * d;
  }
  const float inv = 1.f / (sqrtf(var * (1.f / 34.f)) + 1e-10f);
#pragma unroll
  for (int ch = 0; ch < 35; ++ch)
    spa[(b * 35 + ch) * HW + p] = (v[ch] - mean) * inv;
}

// KernelNorm over the 35 spectral values per (b,c), in place.
__global__ void k_norm_spe(float* __restrict__ spe) {
  const int idx = blockIdx.x * blockDim.x + threadIdx.x;
  if (idx >= BN * CN) return;
  float* row = spe + idx * 35;
  float mean = 0.f;
#pragma unroll
  for (int j = 0; j < 35; ++j) mean += row[j];
  mean *= (1.f / 35.f);
  float var = 0.f;
#pragma unroll
  for (int j = 0; j < 35; ++j) {
    const float d = row[j] - mean;
    var += d * d;
  }
  const float inv = 1.f / (sqrtf(var * (1.f / 34.f)) + 1e-10f);
#pragma unroll
  for (int j = 0; j < 35; ++j) row[j] = (row[j] - mean) * inv;
}

// Final adaptive per-pixel conv: out = sum_{di,dj} x_shift * spe * spa.
template <int K>
__global__ void k_adaptive_conv(const float* __restrict__ xm, const float* __restrict__ spa,
                                const float* __restrict__ spe, float* __restrict__ out,
                                int pad, int off) {
  const int idx = blockIdx.x * blockDim.x + threadIdx.x;
  if (idx >= BN * CN * HW) return;
  const int p  = idx & (HW - 1);
  const int bc = idx >> 14;
  const int c  = bc & 31;
  const int b  = bc >> 5;
  const int yy = p >> 7, xx = p & (WID - 1);
  float sp[K * K];
#pragma unroll
  for (int j = 0; j < K * K; ++j) sp[j] = spe[(b * CN + c) * 35 + off + j];
  float acc = 0.f;
#pragma unroll
  for (int di = 0; di < K; ++di) {
    const int iy = yy + di - pad;
    if (iy < 0 || iy >= WID) continue;
#pragma unroll
    for (int dj = 0; dj < K; ++dj) {
      const int ix = xx + dj - pad;
      if (ix < 0 || ix >= WID) continue;
      const int j = di * K + dj;
      acc += xm[bc * HW + iy * WID + ix] * sp[j] * spa[(b * 35 + off + j) * HW + p];
    }
  }
  out[bc * HW + p] = acc;
}

// ---------------------------------------------------------------------------
extern "C" void kernel_launch(void* const* d_in, const int* in_sizes, int n_in,
                              void* d_out, int out_size, void* d_ws, size_t ws_size,
                              hipStream_t stream) {
  (void)in_sizes; (void)n_in; (void)out_size; (void)ws_size;
  const float* x = (const float*)d_in[0];
  const float* y = (const float*)d_in[1];
  auto P = [&](int i) { return (const float*)d_in[2 + i]; };
  // params order: a0_w,a0_b,a1_w,a1_b,a2_w,a2_b, then per branch i (base 6+16i):
  //  dw1_w,dw1_b,pw1_w,pw1_b,dw2_w,dw2_b,pw2_w,pw2_b,dw3_w,dw3_b,pw3_w,pw3_b,
  //  se1_w,se1_b,se2_w,se2_b

  float* ws   = (float*)d_ws;
  float* bufA = ws;                        // B*C*HW = 2,097,152
  float* bufB = ws + 2097152;              // 2,097,152
  float* xm   = ws + 4194304;              // 2,097,152
  float* ym   = ws + 6291456;              // 2,097,152
  float* spa  = ws + 8388608;              // B*35*HW = 2,293,760
  float* gap  = ws + 10682368;             // 128
  float* g1   = gap + 128;                 // 64
  float* spe  = g1 + 64;                   // B*C*35 = 4480

  const int KS[3]   = {1, 3, 5};
  const int PADS[3] = {0, 1, 2};
  const int OFF[3]  = {0, 1, 10};

  // cross spatial attention
  k_conv1x1_cat_relu<<<4096, 256, 0, stream>>>(x, y, P(0), P(1), bufA);
  k_conv3x3_relu<<<4096, 256, 0, stream>>>(bufA, P(2), P(3), bufB);
  k_attn_apply<<<256, 256, 0, stream>>>(bufB, P(4), P(5), x, y, xm, ym);
  k_gap<<<BN * CN, 256, 0, stream>>>(xm, gap);

  const int nC = BN * CN * HW;  // 2,097,152
  for (int i = 0; i < 3; ++i) {
    const int k = KS[i], pad = PADS[i], k2 = k * k, off = OFF[i];
    const int pb = 6 + i * 16;
    const int nK = BN * k2 * HW;
    // spatial branch: dw -> pw -> dw -> pw(lrelu) -> dw -> pw (into spa slice)
    if (k == 1)      k_dwconv<1><<<(nC + 255) / 256, 256, 0, stream>>>(ym, P(pb + 0), P(pb + 1), bufA, CN, pad);
    else if (k == 3) k_dwconv<3><<<(nC + 255) / 256, 256, 0, stream>>>(ym, P(pb + 0), P(pb + 1), bufA, CN, pad);
    else             k_dwconv<5><<<(nC + 255) / 256, 256, 0, stream>>>(ym, P(pb + 0), P(pb + 1), bufA, CN, pad);
    k_pwconv<<<(nK + 255) / 256, 256, 0, stream>>>(bufA, P(pb + 2), P(pb + 3), bufB, k2, CN, k2 * HW, 0);
    if (k == 1)      k_dwconv<1><<<(nK + 255) / 256, 256, 0, stream>>>(bufB, P(pb + 4), P(pb + 5), bufA, k2, pad);
    else if (k == 3) k_dwconv<3><<<(nK + 255) / 256, 256, 0, stream>>>(bufB, P(pb + 4), P(pb + 5), bufA, k2, pad);
    else             k_dwconv<5><<<(nK + 255) / 256, 256, 0, stream>>>(bufB, P(pb + 4), P(pb + 5), bufA, k2, pad);
    k_pwconv<<<(nK + 255) / 256, 256, 0, stream>>>(bufA, P(pb + 6), P(pb + 7), bufB, k2, k2, k2 * HW, 1);
    if (k == 1)      k_dwconv<1><<<(nK + 255) / 256, 256, 0, stream>>>(bufB, P(pb + 8), P(pb + 9), bufA, k2, pad);
    else if (k == 3) k_dwconv<3><<<(nK + 255) / 256, 256, 0, stream>>>(bufB, P(pb + 8), P(pb + 9), bufA, k2, pad);
    else             k_dwconv<5><<<(nK + 255) / 256, 256, 0, stream>>>(bufB, P(pb + 8), P(pb + 9), bufA, k2, pad);
    k_pwconv<<<(nK + 255) / 256, 256, 0, stream>>>(bufA, P(pb + 10), P(pb + 11), spa + off * HW, k2, k2, 35 * HW, 0);
    // spectral branch
    k_se1<<<1, 64, 0, stream>>>(gap, P(pb + 12), P(pb + 13), g1);
    const int nse2 = BN * CN * k2;
    k_se2<<<(nse2 + 255) / 256, 256, 0, stream>>>(g1, P(pb + 14), P(pb + 15), spe, k2, off);
  }

  k_norm_spa<<<(BN * HW + 255) / 256, 256, 0, stream>>>(spa);
  k_norm_spe<<<1, 128, 0, stream>>>(spe);

  float* out = (float*)d_out;
  k_adaptive_conv<1><<<(nC + 255) / 256, 256, 0, stream>>>(xm, spa, spe, out + 0 * nC, PADS[0], OFF[0]);
  k_adaptive_conv<3><<<(nC + 255) / 256, 256, 0, stream>>>(xm, spa, spe, out + 1 * nC, PADS[1], OFF[1]);
  k_adaptive_conv<5><<<(nC + 255) / 256, 256, 0, stream>>>(xm, spa, spe, out + 2 * nC, PADS[2], OFF[2]);
}